// FapeLoss_12128987644145
// MI455X (gfx1250) — compile-verified
//
#include <hip/hip_runtime.h>
#include <hip/hip_bf16.h>

typedef __attribute__((ext_vector_type(2))) float v2f;
typedef __attribute__((ext_vector_type(8))) float v8f;

#define N_RES   2048
#define M_ROWS  (3 * N_RES)          // 6144 flattened atoms
#define EPSF    0.001f
#define EPMAXF  30.0f
#define ITILES  (M_ROWS / 16)        // 384
#define JTILES  (N_RES / 16)         // 128

// ---------------------------------------------------------------------------
// Stage 1: build GEMM operands in workspace.
//   U[i][8]        = { coor_i(3), tgt_i(3), 1, 0 }                 (6144 x 8)
//   V[e][j][8]     = { p_rot[j,e,:](3), -t_rot[j,e,:](3), b_je, 0 } (3 x 2048 x 8)
// where t_rot/t_tran come from rigid_from_3points(target) and
//   b_je = t_tran_j . t_rot[j,e]  -  p_tran_j . p_rot[j,e]
// ---------------------------------------------------------------------------
__global__ void fape_prep(const float* __restrict__ coor,
                          const float* __restrict__ p_rot,
                          const float* __restrict__ p_tran,
                          const float* __restrict__ target,
                          float* __restrict__ U,
                          float* __restrict__ V) {
    int tid = blockIdx.x * blockDim.x + threadIdx.x;

    if (tid < M_ROWS) {
        float* u = U + tid * 8;
        u[0] = coor[tid * 3 + 0];
        u[1] = coor[tid * 3 + 1];
        u[2] = coor[tid * 3 + 2];
        u[3] = target[tid * 3 + 0];
        u[4] = target[tid * 3 + 1];
        u[5] = target[tid * 3 + 2];
        u[6] = 1.0f;
        u[7] = 0.0f;
    }

    if (tid < N_RES) {
        const float* t = target + tid * 9;
        // x1 = t[0..2], x2 = t[3..5], x3 = t[6..8]
        float x2x = t[3], x2y = t[4], x2z = t[5];
        float v1x = t[6] - x2x, v1y = t[7] - x2y, v1z = t[8] - x2z;
        float v2x = t[0] - x2x, v2y = t[1] - x2y, v2z = t[2] - x2z;

        float n1 = sqrtf(v1x * v1x + v1y * v1y + v1z * v1z) + 0.001f;
        float e1x = v1x / n1, e1y = v1y / n1, e1z = v1z / n1;

        float d = e1x * v2x + e1y * v2y + e1z * v2z;
        float u2x = v2x - e1x * d, u2y = v2y - e1y * d, u2z = v2z - e1z * d;
        float n2 = sqrtf(u2x * u2x + u2y * u2y + u2z * u2z) + 1e-8f;
        float e2x = u2x / n2, e2y = u2y / n2, e2z = u2z / n2;

        float e3x = e1y * e2z - e1z * e2y;
        float e3y = e1z * e2x - e1x * e2z;
        float e3z = e1x * e2y - e1y * e2x;

        float tr[3][3] = { { e1x, e1y, e1z },
                           { e2x, e2y, e2z },
                           { e3x, e3y, e3z } };

        const float* pr = p_rot + tid * 9;
        const float* pt = p_tran + tid * 3;

        #pragma unroll
        for (int e = 0; e < 3; ++e) {
            const float* pre = pr + e * 3;
            float b = (x2x * tr[e][0] + x2y * tr[e][1] + x2z * tr[e][2])
                    - (pt[0] * pre[0] + pt[1] * pre[1] + pt[2] * pre[2]);
            float* vr = V + (size_t)(e * N_RES + tid) * 8;
            vr[0] = pre[0];
            vr[1] = pre[1];
            vr[2] = pre[2];
            vr[3] = -tr[e][0];
            vr[4] = -tr[e][1];
            vr[5] = -tr[e][2];
            vr[6] = b;
            vr[7] = 0.0f;
        }
    }
}

// ---------------------------------------------------------------------------
// Stage 2: the 6144 x 2048 errmap via WMMA f32 16x16x4 (K=8 as two k=4 steps),
// three e-plane accumulators, elementwise sqrt/clip epilogue, block partials.
// Block = one i-tile (16 rows), 4 waves stripe the 128 j-tiles.
// The sqrt argument is >= 3*EP = 0.003 (far from denormals), so we use the
// raw hardware v_sqrt_f32 via __builtin_amdgcn_sqrtf and skip the precise-
// sqrt range-fixup scaffolding (saves ~6 VALU ops per output).
// ---------------------------------------------------------------------------
__global__ void __launch_bounds__(128)
fape_gemm(const float* __restrict__ U,
          const float* __restrict__ V,
          float* __restrict__ partials) {
    const int tid  = threadIdx.x;
    const int wave = tid >> 5;           // 0..3
    const int lane = tid & 31;
    const int n    = lane & 15;          // column / row-in-tile selector
    const int kh   = lane >> 4;          // 0: K pair {0,1}, 1: K pair {2,3}

    const int i0 = blockIdx.x * 16;

    // A operand (shared by every j-tile this wave touches):
    // lanes 0-15 hold {K0,K1} of row M=lane, lanes 16-31 hold {K2,K3}.
    const v2f* urow = (const v2f*)(U + (size_t)(i0 + n) * 8);
    v2f a_lo = urow[kh];        // K 0..3 half
    v2f a_hi = urow[2 + kh];    // K 4..7 half

    float acc = 0.0f;

    for (int jt = wave; jt < JTILES; jt += 4) {
        const int j0 = jt * 16;

        v8f c0 = {}; v8f c1 = {}; v8f c2 = {};

        // e = 0
        {
            const v2f* vrow = (const v2f*)(V + (size_t)(0 * N_RES + j0 + n) * 8);
            v2f b_lo = vrow[kh];
            v2f b_hi = vrow[2 + kh];
            c0 = __builtin_amdgcn_wmma_f32_16x16x4_f32(false, a_lo, false, b_lo,
                                                       (short)0, c0, false, false);
            c0 = __builtin_amdgcn_wmma_f32_16x16x4_f32(false, a_hi, false, b_hi,
                                                       (short)0, c0, false, false);
        }
        // e = 1
        {
            const v2f* vrow = (const v2f*)(V + (size_t)(1 * N_RES + j0 + n) * 8);
            v2f b_lo = vrow[kh];
            v2f b_hi = vrow[2 + kh];
            c1 = __builtin_amdgcn_wmma_f32_16x16x4_f32(false, a_lo, false, b_lo,
                                                       (short)0, c1, false, false);
            c1 = __builtin_amdgcn_wmma_f32_16x16x4_f32(false, a_hi, false, b_hi,
                                                       (short)0, c1, false, false);
        }
        // e = 2
        {
            const v2f* vrow = (const v2f*)(V + (size_t)(2 * N_RES + j0 + n) * 8);
            v2f b_lo = vrow[kh];
            v2f b_hi = vrow[2 + kh];
            c2 = __builtin_amdgcn_wmma_f32_16x16x4_f32(false, a_lo, false, b_lo,
                                                       (short)0, c2, false, false);
            c2 = __builtin_amdgcn_wmma_f32_16x16x4_f32(false, a_hi, false, b_hi,
                                                       (short)0, c2, false, false);
        }

        // Epilogue: each of the 8 accumulator regs in each of the 3 planes
        // holds the same (i,j) pair across planes -> pure elementwise math.
        #pragma unroll
        for (int v = 0; v < 8; ++v) {
            float s = fmaf(c0[v], c0[v], 3.0f * EPSF);
            s = fmaf(c1[v], c1[v], s);
            s = fmaf(c2[v], c2[v], s);
            float r = __builtin_amdgcn_sqrtf(s);   // raw v_sqrt_f32
            acc += fminf(r, EPMAXF);
        }
    }

    // Deterministic block reduction.
    __shared__ float red[128];
    red[tid] = acc;
    __syncthreads();
    #pragma unroll
    for (int off = 64; off > 0; off >>= 1) {
        if (tid < off) red[tid] += red[tid + off];
        __syncthreads();
    }
    if (tid == 0) partials[blockIdx.x] = red[0];
}

// ---------------------------------------------------------------------------
// Stage 3: fixed-order reduction of the 384 block partials -> scalar loss.
// ---------------------------------------------------------------------------
__global__ void fape_finish(const float* __restrict__ partials,
                            float* __restrict__ out) {
    __shared__ float red[128];
    int t = threadIdx.x;
    float s = partials[t] + partials[t + 128] + partials[t + 256];
    red[t] = s;
    __syncthreads();
    #pragma unroll
    for (int off = 64; off > 0; off >>= 1) {
        if (t < off) red[t] += red[t + off];
        __syncthreads();
    }
    if (t == 0) {
        const float scale = 1.0f / (6144.0f * 2048.0f * 10.0f);
        out[0] = red[0] * scale;
    }
}

// ---------------------------------------------------------------------------
extern "C" void kernel_launch(void* const* d_in, const int* in_sizes, int n_in,
                              void* d_out, int out_size, void* d_ws, size_t ws_size,
                              hipStream_t stream) {
    const float* coor   = (const float*)d_in[0];   // (2048,3,3)
    const float* p_rot  = (const float*)d_in[1];   // (2048,3,3)
    const float* p_tran = (const float*)d_in[2];   // (2048,3)
    const float* target = (const float*)d_in[3];   // (2048,3,3)
    float* out = (float*)d_out;

    float* ws = (float*)d_ws;
    float* U        = ws;                              // 6144*8 floats
    float* V        = ws + (size_t)M_ROWS * 8;         // 3*2048*8 floats
    float* partials = V + (size_t)3 * N_RES * 8;       // 384 floats

    fape_prep  <<<(M_ROWS + 255) / 256, 256, 0, stream>>>(coor, p_rot, p_tran, target, U, V);
    fape_gemm  <<<ITILES, 128, 0, stream>>>(U, V, partials);
    fape_finish<<<1, 128, 0, stream>>>(partials, out);
}